// B_29729763623191
// MI455X (gfx1250) — compile-verified
//
#include <hip/hip_runtime.h>
#include <math.h>

#define CDIM   64
#define HDIM   192
#define WDIM   192
#define BATCHN 4
#define HW     (HDIM * WDIM)        /* 36864 */
#define HEADS  2

static const long CHW_L  = (long)CDIM * HW;       /* 2359296 */
static const long C3HW_L = 3L * CDIM * HW;        /* 7077888 */

typedef __attribute__((ext_vector_type(2))) float v2f;
typedef __attribute__((ext_vector_type(8))) float v8f;

__device__ __forceinline__ v8f wmma4(v2f a, v2f b, v8f c) {
  // D(16x16,f32) += A(16x4,f32) * B(4x16,f32), full fp32 precision
  return __builtin_amdgcn_wmma_f32_16x16x4_f32(false, a, false, b, (short)0, c,
                                               false, false);
}

__device__ __forceinline__ float gelu_erf(float x) {
  return 0.5f * x * (1.0f + erff(x * 0.70710678118654752440f));
}

// ---------------------------------------------------------------------------
// Generic 1x1-conv GEMM:  Y[b][m][p] = actO( sum_k W[b][m][k]*X[b][k][p]
//                                            + bias[m] + Add[b][m][p] )
// One wave computes a 16(M) x 64(N) strip (4 WMMA accumulators, A reused 4x).
// blockDim.x = (M/16)*32 ; grid = (HW/64, nbatches).
// ---------------------------------------------------------------------------
__global__ void k_conv1x1(const float* __restrict__ X, const float* __restrict__ Wt,
                          const float* __restrict__ bias, const float* __restrict__ add,
                          float* __restrict__ Y,
                          int Kdim, long in_bs, long out_bs, long add_bs, long w_bs,
                          int gelu_out)
{
  const int ng   = blockIdx.x;      // group of 64 output columns
  const int bb   = blockIdx.y;
  const int wave = threadIdx.x >> 5;
  const int lane = threadIdx.x & 31;
  const int l16  = lane & 15;
  const int half = lane >> 4;

  const float* Xb = X  + (long)bb * in_bs;
  const float* Wb = Wt + (long)bb * w_bs;
  const int m  = wave * 16 + l16;   // A-frag row (M index)
  const int n0 = ng * 64 + l16;     // first B-frag col

  v8f acc0 = {}, acc1 = {}, acc2 = {}, acc3 = {};
  for (int kk = 0; kk < Kdim; kk += 4) {
    const int k0 = kk + half * 2;
    v2f a;
    a.x = Wb[m * Kdim + k0];
    a.y = Wb[m * Kdim + k0 + 1];
    const float* xp0 = Xb + (long)k0 * HW + n0;
    const float* xp1 = xp0 + HW;
    v2f b0; b0.x = xp0[0];  b0.y = xp1[0];
    v2f b1; b1.x = xp0[16]; b1.y = xp1[16];
    v2f b2; b2.x = xp0[32]; b2.y = xp1[32];
    v2f b3; b3.x = xp0[48]; b3.y = xp1[48];
    acc0 = wmma4(a, b0, acc0);
    acc1 = wmma4(a, b1, acc1);
    acc2 = wmma4(a, b2, acc2);
    acc3 = wmma4(a, b3, acc3);
  }

  float* Yb = Y + (long)bb * out_bs;
  const float* Ab = add ? (add + (long)bb * add_bs) : nullptr;
#pragma unroll
  for (int t = 0; t < 4; ++t) {
    const v8f acc = (t == 0) ? acc0 : (t == 1) ? acc1 : (t == 2) ? acc2 : acc3;
    const int n = n0 + 16 * t;
#pragma unroll
    for (int v = 0; v < 8; ++v) {
      const int row = wave * 16 + v + half * 8;
      float val = acc[v];
      if (bias) val += bias[row];
      if (Ab)   val += Ab[(long)row * HW + n];
      if (gelu_out) val = gelu_erf(val);
      Yb[(long)row * HW + n] = val;
    }
  }
}

// ---------------------------------------------------------------------------
// Generic per-pair strided GEMM (row/col attention scores + apply).
//  pair base:  (p/pdiv)*X_bs + (p%pdiv)*X_is  for A, B, Out, Res
//  D[m,n] = sum_k A[m*a_ms + k*a_ks] * B[k*b_ks + n*b_ns]
//  if Res: Out = gamma*D + Res  else Out = D
// Ndim must be a multiple of 64. One wave = one 16x64 strip; 4 waves/block.
// ---------------------------------------------------------------------------
__global__ void k_gemm_pair(const float* __restrict__ A, const float* __restrict__ Bm,
                            const float* __restrict__ Res, const float* __restrict__ gammap,
                            float* __restrict__ Out,
                            int Mdim, int Ndim, int Kdim, int pdiv,
                            long a_bs, long a_is, long a_ms, long a_ks,
                            long b_bs, long b_is, long b_ks, long b_ns,
                            long o_bs, long o_is, long o_ms, long o_ns,
                            long r_bs, long r_is)
{
  const int pair = blockIdx.y;
  const int wave = threadIdx.x >> 5;
  const int lane = threadIdx.x & 31;
  const int l16  = lane & 15;
  const int half = lane >> 4;
  const int ngs  = Ndim >> 6;                 // strips of 64 along N
  const int mt   = Mdim >> 4;
  const int strip = blockIdx.x * 4 + wave;    // wave-uniform
  if (strip >= mt * ngs) return;
  const int mtile = strip / ngs;
  const int ng    = strip % ngs;

  const long pa = (long)(pair / pdiv) * a_bs + (long)(pair % pdiv) * a_is;
  const long pb = (long)(pair / pdiv) * b_bs + (long)(pair % pdiv) * b_is;
  const long po = (long)(pair / pdiv) * o_bs + (long)(pair % pdiv) * o_is;

  const int m  = mtile * 16 + l16;
  const int n0 = ng * 64 + l16;

  v8f acc0 = {}, acc1 = {}, acc2 = {}, acc3 = {};
  for (int kk = 0; kk < Kdim; kk += 4) {
    const int k0 = kk + half * 2;
    v2f a;
    a.x = A[pa + (long)m * a_ms + (long)k0 * a_ks];
    a.y = A[pa + (long)m * a_ms + (long)(k0 + 1) * a_ks];
    const long base0 = pb + (long)k0 * b_ks;
    const long base1 = base0 + b_ks;
    v2f b0; b0.x = Bm[base0 + (long)(n0     ) * b_ns]; b0.y = Bm[base1 + (long)(n0     ) * b_ns];
    v2f b1; b1.x = Bm[base0 + (long)(n0 + 16) * b_ns]; b1.y = Bm[base1 + (long)(n0 + 16) * b_ns];
    v2f b2; b2.x = Bm[base0 + (long)(n0 + 32) * b_ns]; b2.y = Bm[base1 + (long)(n0 + 32) * b_ns];
    v2f b3; b3.x = Bm[base0 + (long)(n0 + 48) * b_ns]; b3.y = Bm[base1 + (long)(n0 + 48) * b_ns];
    acc0 = wmma4(a, b0, acc0);
    acc1 = wmma4(a, b1, acc1);
    acc2 = wmma4(a, b2, acc2);
    acc3 = wmma4(a, b3, acc3);
  }

  const float g = Res ? gammap[0] : 1.0f;
  const long pr = Res ? ((long)(pair / pdiv) * r_bs + (long)(pair % pdiv) * r_is) : 0;
#pragma unroll
  for (int t = 0; t < 4; ++t) {
    const v8f acc = (t == 0) ? acc0 : (t == 1) ? acc1 : (t == 2) ? acc2 : acc3;
    const int n = n0 + 16 * t;
#pragma unroll
    for (int v = 0; v < 8; ++v) {
      const int row = mtile * 16 + v + half * 8;
      float val = acc[v];
      if (Res) val = g * val + Res[pr + (long)row * o_ms + (long)n * o_ns];
      Out[po + (long)row * o_ms + (long)n * o_ns] = val;
    }
  }
}

// ---------------------------------------------------------------------------
// Depthwise 3x3, padding 1, + bias, optional erf-GELU on the output.
// ---------------------------------------------------------------------------
__global__ void k_dwconv3x3(const float* __restrict__ X, const float* __restrict__ Wd,
                            const float* __restrict__ Bd, float* __restrict__ Y,
                            int apply_gelu)
{
  long idx = (long)blockIdx.x * blockDim.x + threadIdx.x;
  if (idx >= (long)BATCHN * CDIM * HW) return;
  const int p = (int)(idx % HW);
  const long bc = idx / HW;
  const int c = (int)(bc % CDIM);
  const int y = p / WDIM, x = p % WDIM;
  const float* Xp = X + bc * HW;
  float s = Bd[c];
#pragma unroll
  for (int dy = 0; dy < 3; ++dy) {
    const int yy = y + dy - 1;
    if (yy < 0 || yy >= HDIM) continue;
#pragma unroll
    for (int dx = 0; dx < 3; ++dx) {
      const int xx = x + dx - 1;
      if (xx < 0 || xx >= WDIM) continue;
      s += Wd[c * 9 + dy * 3 + dx] * Xp[yy * WDIM + xx];
    }
  }
  if (apply_gelu) s = gelu_erf(s);
  Y[idx] = s;
}

// ---------------------------------------------------------------------------
// Per-row inverse L2 norm: scale[r] = 1/max(||X[r,:]||, 1e-12), n elements.
// ---------------------------------------------------------------------------
__global__ void k_rownorm(const float* __restrict__ X, float* __restrict__ scale, int n)
{
  __shared__ float red[256];
  const float* p = X + (long)blockIdx.x * n;
  float s = 0.0f;
  for (int i = threadIdx.x; i < n; i += 256) { const float v = p[i]; s += v * v; }
  red[threadIdx.x] = s;
  __syncthreads();
  for (int off = 128; off > 0; off >>= 1) {
    if ((int)threadIdx.x < off) red[threadIdx.x] += red[threadIdx.x + off];
    __syncthreads();
  }
  if (threadIdx.x == 0) scale[blockIdx.x] = 1.0f / fmaxf(sqrtf(red[0]), 1e-12f);
}

// ---------------------------------------------------------------------------
// Channel-attention logits: attn[bh,c,d] = (q_c . k_d) * sq[c]*sk[d]*temp[head]
// One block per (bh,c,d); grid = 8*32*32 = 8192.
// ---------------------------------------------------------------------------
__global__ void k_attn_dot(const float* __restrict__ q, const float* __restrict__ k,
                           const float* __restrict__ sq, const float* __restrict__ sk,
                           const float* __restrict__ temp, float* __restrict__ attn, int n)
{
  __shared__ float red[256];
  const int idx = blockIdx.x;
  const int bh = idx >> 10;
  const int c  = (idx >> 5) & 31;
  const int d  = idx & 31;
  const float* qp = q + ((long)bh * 32 + c) * n;
  const float* kp = k + ((long)bh * 32 + d) * n;
  float s = 0.0f;
  for (int i = threadIdx.x; i < n; i += 256) s += qp[i] * kp[i];
  red[threadIdx.x] = s;
  __syncthreads();
  for (int off = 128; off > 0; off >>= 1) {
    if ((int)threadIdx.x < off) red[threadIdx.x] += red[threadIdx.x + off];
    __syncthreads();
  }
  if (threadIdx.x == 0)
    attn[idx] = red[0] * sq[bh * 32 + c] * sk[bh * 32 + d] * temp[bh & (HEADS - 1)];
}

// Softmax over rows of 32 (channel attention). 8 waves/block, wave per row.
__global__ void k_softmax32(float* __restrict__ attn)
{
  const int wave = threadIdx.x >> 5, lane = threadIdx.x & 31;
  const int row = blockIdx.x * 8 + wave;
  float* p = attn + (long)row * 32;
  float v = p[lane];
  float mx = v;
  for (int o = 16; o > 0; o >>= 1) mx = fmaxf(mx, __shfl_xor(mx, o, 32));
  const float e = expf(v - mx);
  float s = e;
  for (int o = 16; o > 0; o >>= 1) s += __shfl_xor(s, o, 32);
  p[lane] = e / s;
}

// Softmax over rows of 192 (row/col attention). 8 waves/block, wave per row.
__global__ void k_softmax192(float* __restrict__ S)
{
  const int wave = threadIdx.x >> 5, lane = threadIdx.x & 31;
  const long row = (long)blockIdx.x * 8 + wave;
  float* p = S + row * 192;
  float v[6];
  float mx = -INFINITY;
#pragma unroll
  for (int t = 0; t < 6; ++t) { v[t] = p[lane + 32 * t]; mx = fmaxf(mx, v[t]); }
  for (int o = 16; o > 0; o >>= 1) mx = fmaxf(mx, __shfl_xor(mx, o, 32));
  float s = 0.0f;
#pragma unroll
  for (int t = 0; t < 6; ++t) { v[t] = expf(v[t] - mx); s += v[t]; }
  for (int o = 16; o > 0; o >>= 1) s += __shfl_xor(s, o, 32);
  const float inv = 1.0f / s;
#pragma unroll
  for (int t = 0; t < 6; ++t) p[lane + 32 * t] = v[t] * inv;
}

// ---------------------------------------------------------------------------
extern "C" void kernel_launch(void* const* d_in, const int* in_sizes, int n_in,
                              void* d_out, int out_size, void* d_ws, size_t ws_size,
                              hipStream_t stream) {
  const float* x        = (const float*)d_in[0];
  const float* pw_w     = (const float*)d_in[1];
  const float* dw_w     = (const float*)d_in[2];
  const float* dw_b     = (const float*)d_in[3];
  const float* conv2_w  = (const float*)d_in[4];
  const float* conv2_b  = (const float*)d_in[5];
  const float* conv0_w  = (const float*)d_in[6];
  const float* conv0_b  = (const float*)d_in[7];
  const float* att_q_w  = (const float*)d_in[8];
  const float* att_k_w  = (const float*)d_in[9];
  const float* att_v_w  = (const float*)d_in[10];
  const float* att_p_w  = (const float*)d_in[11];
  const float* temp     = (const float*)d_in[12];
  const float* row_q_w  = (const float*)d_in[13];
  const float* row_k_w  = (const float*)d_in[14];
  const float* row_v_w  = (const float*)d_in[15];
  const float* row_g    = (const float*)d_in[16];
  const float* col_q_w  = (const float*)d_in[17];
  const float* col_k_w  = (const float*)d_in[18];
  const float* col_v_w  = (const float*)d_in[19];
  const float* col_g    = (const float*)d_in[20];
  const float* conv_w   = (const float*)d_in[21];
  const float* conv_b   = (const float*)d_in[22];
  float* out = (float*)d_out;

  const long S = (long)BATCHN * CDIM * HW;   // 9,437,184 floats per slot
  float* ws   = (float*)d_ws;
  float* bufA = ws;            // t  / q    / rowQ / colQ
  float* bufB = ws + S;        // gelu(t2) / k / rowK / colK
  float* bufC = ws + 2 * S;    // conv0-out / v / rowV / colV
  float* x1   = ws + 3 * S;
  float* bufE = ws + 4 * S;    // out1 (pre-proj)
  float* cat  = ws + 5 * S;    // [4][192][HW]  (3 slots)
  float* G    = ws + 8 * S;    // scores [768][192][192] (3 slots)
  float* scq  = ws + 11 * S;   // 256
  float* sck  = scq + 256;     // 256
  float* attn = sck + 256;     // 8*32*32 = 8192

  const dim3 g1(HW / 64, BATCHN), b128(128);
  const int PAIRS = BATCHN * HDIM;   // 768

  // ---- t = pw(x) ----
  k_conv1x1<<<g1, b128, 0, stream>>>(x, pw_w, nullptr, nullptr, bufA,
                                     CDIM, CHW_L, CHW_L, 0, 0, 0);
  // ---- bufB = gelu(dw3x3(t) + dw_b) ----
  k_dwconv3x3<<<dim3((unsigned)((S + 255) / 256)), dim3(256), 0, stream>>>(
      bufA, dw_w, dw_b, bufB, 1);
  // ---- bufC = conv0(x) + b ----
  k_conv1x1<<<g1, b128, 0, stream>>>(x, conv0_w, conv0_b, nullptr, bufC,
                                     CDIM, CHW_L, CHW_L, 0, 0, 0);
  // ---- x1 = conv2(gelu_t2) + b + bufC ----
  k_conv1x1<<<g1, b128, 0, stream>>>(bufB, conv2_w, conv2_b, bufC, x1,
                                     CDIM, CHW_L, CHW_L, CHW_L, 0, 0);

  // ---- channel attention ----
  k_conv1x1<<<g1, b128, 0, stream>>>(x1, att_q_w, nullptr, nullptr, bufA,
                                     CDIM, CHW_L, CHW_L, 0, 0, 0);
  k_conv1x1<<<g1, b128, 0, stream>>>(x1, att_k_w, nullptr, nullptr, bufB,
                                     CDIM, CHW_L, CHW_L, 0, 0, 0);
  k_conv1x1<<<g1, b128, 0, stream>>>(x1, att_v_w, nullptr, nullptr, bufC,
                                     CDIM, CHW_L, CHW_L, 0, 0, 0);
  k_rownorm<<<dim3(BATCHN * CDIM), dim3(256), 0, stream>>>(bufA, scq, HW);
  k_rownorm<<<dim3(BATCHN * CDIM), dim3(256), 0, stream>>>(bufB, sck, HW);
  k_attn_dot<<<dim3(BATCHN * HEADS * 32 * 32), dim3(256), 0, stream>>>(
      bufA, bufB, scq, sck, temp, attn, HW);
  k_softmax32<<<dim3(32), dim3(256), 0, stream>>>(attn);
  // out1 = attn @ v :  8 "batches" (b,head), per-batch 32x32 weights
  k_conv1x1<<<dim3(HW / 64, BATCHN * HEADS), dim3(64), 0, stream>>>(
      bufC, attn, nullptr, nullptr, bufE,
      32, 32L * HW, 32L * HW, 0, 32L * 32, 0);
  // proj -> cat slice 0
  k_conv1x1<<<g1, b128, 0, stream>>>(bufE, att_p_w, nullptr, nullptr, cat,
                                     CDIM, CHW_L, C3HW_L, 0, 0, 0);

  // ---- row attention ----
  k_conv1x1<<<g1, b128, 0, stream>>>(x1, row_q_w, nullptr, nullptr, bufA,
                                     CDIM, CHW_L, CHW_L, 0, 0, 0);
  k_conv1x1<<<g1, b128, 0, stream>>>(x1, row_k_w, nullptr, nullptr, bufB,
                                     CDIM, CHW_L, CHW_L, 0, 0, 0);
  k_conv1x1<<<g1, b128, 0, stream>>>(cat, row_v_w, nullptr, nullptr, bufC,
                                     CDIM, C3HW_L, CHW_L, 0, 0, 0);
  // scores: s[p][i][j] = sum_c Q[c, h*W+i] K[c, h*W+j], p = b*192+h
  k_gemm_pair<<<dim3(9, PAIRS), b128, 0, stream>>>(
      bufA, bufB, nullptr, nullptr, G,
      192, 192, CDIM, HDIM,
      CHW_L, (long)WDIM, 1L, (long)HW,        // A: base b*CHW + h*W, m=i (1), k=c (HW)
      CHW_L, (long)WDIM, (long)HW, 1L,        // B: k=c (HW), n=j (1)
      192L * HW, (long)HW, 192L, 1L,          // Out = G, pair*36864
      0, 0);
  k_softmax192<<<dim3(PAIRS * HDIM / 8), dim3(256), 0, stream>>>(G);
  // out2 = gamma * (V . a^T) + x1 -> cat slice 1
  k_gemm_pair<<<dim3(3, PAIRS), b128, 0, stream>>>(
      bufC, G, x1, row_g, cat + 64L * HW,
      CDIM, 192, 192, HDIM,
      CHW_L, (long)WDIM, (long)HW, 1L,        // A = V: m=c (HW), k=j (1)
      192L * HW, (long)HW, 1L, 192L,          // B = a: a[i,j] -> k=j (1), n=i (192)
      C3HW_L, (long)WDIM, (long)HW, 1L,       // Out: cat slice1
      CHW_L, (long)WDIM);                     // Res = x1

  // ---- col attention ----
  k_conv1x1<<<g1, b128, 0, stream>>>(x1, col_q_w, nullptr, nullptr, bufA,
                                     CDIM, CHW_L, CHW_L, 0, 0, 0);
  k_conv1x1<<<g1, b128, 0, stream>>>(x1, col_k_w, nullptr, nullptr, bufB,
                                     CDIM, CHW_L, CHW_L, 0, 0, 0);
  k_conv1x1<<<g1, b128, 0, stream>>>(cat, col_v_w, nullptr, nullptr, bufC,
                                     CDIM, C3HW_L, CHW_L, 0, 0, 0);
  // scores: s[p][i][j] = sum_c Q[c, i*W+w] K[c, j*W+w], p = b*192+w
  k_gemm_pair<<<dim3(9, PAIRS), b128, 0, stream>>>(
      bufA, bufB, nullptr, nullptr, G,
      192, 192, CDIM, HDIM,
      CHW_L, 1L, (long)WDIM, (long)HW,        // A: base b*CHW + w, m=i (W), k=c (HW)
      CHW_L, 1L, (long)HW, (long)WDIM,        // B: k=c (HW), n=j (W)
      192L * HW, (long)HW, 192L, 1L,
      0, 0);
  k_softmax192<<<dim3(PAIRS * HDIM / 8), dim3(256), 0, stream>>>(G);
  // out3 = gamma * (V . a^T) + x1 -> cat slice 2
  k_gemm_pair<<<dim3(3, PAIRS), b128, 0, stream>>>(
      bufC, G, x1, col_g, cat + 128L * HW,
      CDIM, 192, 192, HDIM,
      CHW_L, 1L, (long)HW, (long)WDIM,        // A = V: m=c (HW), k=j (W)
      192L * HW, (long)HW, 1L, 192L,          // B = a: k=j (1), n=i (192)
      C3HW_L, 1L, (long)HW, (long)WDIM,       // Out: cat slice2, m=c (HW), n=i (W)
      CHW_L, 1L);                             // Res = x1

  // ---- final fuse: gelu(conv_w @ cat + conv_b) ----
  k_conv1x1<<<g1, b128, 0, stream>>>(cat, conv_w, conv_b, nullptr, out,
                                     3 * CDIM, C3HW_L, CHW_L, 0, 0, 1);
}